// SuperRGCNLayer_86835648791384
// MI455X (gfx1250) — compile-verified
//
#include <hip/hip_runtime.h>

// SuperRGCNLayer for MI455X (gfx1250, wave32).
// T=2 node types, R=4 rel types, M=16 distinct 64x64 matrices, B=8 bases.
// Strategy: counting-sort edges by relation id (16 bins), then each wave does a
// 16-edge x (64x64) GEMM tile with V_WMMA_F32_16X16X4_F32 and scatters with
// global f32 atomics. x (25.6MB) and out (25.6MB) are L2-resident on MI455X.

#define NB     8
#define INDIM  64
#define OUTDIM 64
#define MREL   16

typedef __attribute__((ext_vector_type(2))) float v2f;
typedef __attribute__((ext_vector_type(8))) float v8f;

// ---------------------------------------------------------------- small utils
__global__ void k_zero_counters(int* cnt, int* cursor) {
  int t = threadIdx.x;
  if (t < MREL) { cnt[t] = 0; cursor[t] = 0; }
}

__global__ void k_bias_init(float* __restrict__ out, const float* __restrict__ bias, int total) {
  for (int i = blockIdx.x * blockDim.x + threadIdx.x; i < total; i += gridDim.x * blockDim.x)
    out[i] = bias[i & (OUTDIM - 1)];
}

// Wt[m][i][o] = W[m][o][i] = sum_b w_comp[o%16][b] * weightFlat[(m*4 + o/16)*512 + b*64 + i]
// (faithful to the torch .view raw reinterpretations in the reference)
__global__ void k_compute_wt(float* __restrict__ Wt, const float* __restrict__ weight,
                             const float* __restrict__ w_comp) {
  int tid = blockIdx.x * blockDim.x + threadIdx.x;
  if (tid >= MREL * INDIM * OUTDIM) return;
  int m = tid >> 12, rem = tid & 4095, i = rem >> 6, o = rem & 63;
  const float* wrow = weight + (m * 4 + (o >> 4)) * 512 + i;
  const float* cr   = w_comp + (o & 15) * NB;
  float acc = 0.f;
#pragma unroll
  for (int b = 0; b < NB; ++b) acc += cr[b] * wrow[b * 64];
  Wt[tid] = acc;
}

__device__ __forceinline__ int rel_of(const int* nt, const int* es, const int* ed,
                                      const int* et, int e) {
  return nt[es[e]] * 8 + nt[ed[e]] * 4 + et[e];
}

__global__ void k_hist(const int* __restrict__ nt, const int* __restrict__ es,
                       const int* __restrict__ ed, const int* __restrict__ et,
                       int* __restrict__ cnt, int E) {
  for (int e = blockIdx.x * blockDim.x + threadIdx.x; e < E; e += gridDim.x * blockDim.x)
    atomicAdd(&cnt[rel_of(nt, es, ed, et, e)], 1);
}

__global__ void k_scan(const int* __restrict__ cnt, int* __restrict__ binStart,
                       int* __restrict__ cursor) {
  if (threadIdx.x == 0) {
    int run = 0;
    for (int m = 0; m < MREL; ++m) { binStart[m] = run; cursor[m] = run; run += cnt[m]; }
  }
}

__global__ void k_scatter_idx(const int* __restrict__ nt, const int* __restrict__ es,
                              const int* __restrict__ ed, const int* __restrict__ et,
                              int* __restrict__ cursor, int* __restrict__ order, int E) {
  for (int e = blockIdx.x * blockDim.x + threadIdx.x; e < E; e += gridDim.x * blockDim.x) {
    int r = rel_of(nt, es, ed, et, e);
    order[atomicAdd(&cursor[r], 1)] = e;
  }
}

// ---------------------------------------------------------------- main kernel
// 128 threads = 4 waves per workgroup. One wave handles a tile of 16 edges of a
// single relation m: C(16x64) = A(16x64) * B(64x64), B = Wt[m] staged in LDS.
// K-loop of 16 steps of V_WMMA_F32_16X16X4_F32 x 4 output n-tiles.
__global__ __launch_bounds__(128)
void k_main(const float* __restrict__ x, const int* __restrict__ es,
            const int* __restrict__ ed, const int* __restrict__ order,
            const float* __restrict__ Wt, const int* __restrict__ cnt,
            const int* __restrict__ binStart, float* __restrict__ out) {
  __shared__ float sB[INDIM * OUTDIM];  // 16 KB: Wt[m], layout [i][o]

  const int lane = threadIdx.x & 31;
  const int gw   = blockIdx.x * (blockDim.x >> 5) + (threadIdx.x >> 5);
  const int nw   = gridDim.x * (blockDim.x >> 5);
  const int slot  = lane & 15;          // which of the 16 edges this lane serves
  const int khalf = (lane >> 4) << 1;   // 0 or 2: A/B fragment K sub-offset
  const int oc    = lane & 15;          // out-channel within an n-tile

  for (int m = 0; m < MREL; ++m) {
    // stage B = Wt[m] into LDS (whole workgroup)
    for (int t = threadIdx.x; t < INDIM * OUTDIM; t += blockDim.x)
      sB[t] = Wt[m * INDIM * OUTDIM + t];
    __syncthreads();

    const int c  = cnt[m];
    const int bs = binStart[m];
    const int ntiles = (c + 15) >> 4;

    for (int w = gw; w < ntiles; w += nw) {
      const int base = bs + (w << 4);
      int nval = c - (w << 4); if (nval > 16) nval = 16;

      const int eid = order[base + (slot < nval ? slot : nval - 1)];
      const int src = es[eid];
      const int dst = ed[eid];
      const float* xr = x + src * INDIM;

      v8f acc0 = {}; v8f acc1 = {}; v8f acc2 = {}; v8f acc3 = {};

#pragma unroll
      for (int k0 = 0; k0 < INDIM; k0 += 4) {
        // A fragment 16x4 f32: lane L holds row M=L%16, K = k0 + 2*(L/16) + {0,1}
        v2f a = *(const v2f*)(xr + k0 + khalf);
        // B fragments 4x16: lane L reg v holds K = k0 + 2*(L/16) + v, N = n0 + L%16
        const float* bp = sB + (k0 + khalf) * OUTDIM + oc;
        v2f b0 = { bp[0],      bp[OUTDIM]      };
        v2f b1 = { bp[16],     bp[OUTDIM + 16] };
        v2f b2 = { bp[32],     bp[OUTDIM + 32] };
        v2f b3 = { bp[48],     bp[OUTDIM + 48] };
        acc0 = __builtin_amdgcn_wmma_f32_16x16x4_f32(false, a, false, b0, (short)0, acc0, false, false);
        acc1 = __builtin_amdgcn_wmma_f32_16x16x4_f32(false, a, false, b1, (short)0, acc1, false, false);
        acc2 = __builtin_amdgcn_wmma_f32_16x16x4_f32(false, a, false, b2, (short)0, acc2, false, false);
        acc3 = __builtin_amdgcn_wmma_f32_16x16x4_f32(false, a, false, b3, (short)0, acc3, false, false);
      }

      // C layout: lane L, VGPR j -> row M = j + 8*(L/16) (edge slot), col N = oc.
#pragma unroll
      for (int j = 0; j < 8; ++j) {
        const int e = j + ((lane >> 4) << 3);
        const int d = __shfl(dst, e, 32);     // dst node of edge slot e
        if (e < nval) {
          float* op = out + d * OUTDIM + oc;
          atomicAdd(op + 0,  acc0[j]);
          atomicAdd(op + 16, acc1[j]);
          atomicAdd(op + 32, acc2[j]);
          atomicAdd(op + 48, acc3[j]);
        }
      }
    }
    __syncthreads();
  }
}

// ------------------------------------------------- fallback (no scratch path)
__global__ void k_fallback(const float* __restrict__ x, const int* __restrict__ nt,
                           const int* __restrict__ es, const int* __restrict__ ed,
                           const int* __restrict__ et, const float* __restrict__ weight,
                           const float* __restrict__ w_comp, float* __restrict__ out, int E) {
  const int total = E * OUTDIM;
  for (int idx = blockIdx.x * blockDim.x + threadIdx.x; idx < total;
       idx += gridDim.x * blockDim.x) {
    const int e = idx >> 6, o = idx & 63;
    const int r = nt[es[e]] * 8 + nt[ed[e]] * 4 + et[e];
    const float* xr = x + es[e] * INDIM;
    const float* wb = weight + (r * 4 + (o >> 4)) * 512;
    float acc = 0.f;
    for (int b = 0; b < NB; ++b) {
      float dot = 0.f;
      for (int i = 0; i < INDIM; ++i) dot += wb[b * 64 + i] * xr[i];
      acc += w_comp[(o & 15) * NB + b] * dot;
    }
    atomicAdd(&out[ed[e] * OUTDIM + o], acc);
  }
}

// ---------------------------------------------------------------------- entry
extern "C" void kernel_launch(void* const* d_in, const int* in_sizes, int n_in,
                              void* d_out, int out_size, void* d_ws, size_t ws_size,
                              hipStream_t stream) {
  const float* x         = (const float*)d_in[0];
  const int*   node_type = (const int*)d_in[1];
  const int*   edge_src  = (const int*)d_in[2];
  const int*   edge_dst  = (const int*)d_in[3];
  const int*   edge_type = (const int*)d_in[4];
  const float* weight    = (const float*)d_in[5];
  const float* w_comp    = (const float*)d_in[6];
  const float* bias      = (const float*)d_in[7];
  float* out = (float*)d_out;

  const int N = in_sizes[0] / INDIM;
  const int E = in_sizes[2];

  // workspace layout
  const size_t offWt     = 0;                          // 65536 f32 = 256 KB
  const size_t offCnt    = offWt + (size_t)MREL * INDIM * OUTDIM * 4;
  const size_t offStart  = offCnt + 64;
  const size_t offCursor = offStart + 64;
  const size_t offOrder  = (offCursor + 64 + 255) & ~(size_t)255;
  const size_t need      = offOrder + (size_t)E * 4;

  // out = bias (broadcast), then accumulate messages
  k_bias_init<<<2048, 256, 0, stream>>>(out, bias, N * OUTDIM);

  if (ws_size >= need) {
    float* Wt       = (float*)((char*)d_ws + offWt);
    int*   cnt      = (int*)((char*)d_ws + offCnt);
    int*   binStart = (int*)((char*)d_ws + offStart);
    int*   cursor   = (int*)((char*)d_ws + offCursor);
    int*   order    = (int*)((char*)d_ws + offOrder);

    k_zero_counters<<<1, 32, 0, stream>>>(cnt, cursor);
    k_compute_wt<<<(MREL * INDIM * OUTDIM + 255) / 256, 256, 0, stream>>>(Wt, weight, w_comp);
    k_hist<<<1024, 256, 0, stream>>>(node_type, edge_src, edge_dst, edge_type, cnt, E);
    k_scan<<<1, 32, 0, stream>>>(cnt, binStart, cursor);
    k_scatter_idx<<<1024, 256, 0, stream>>>(node_type, edge_src, edge_dst, edge_type,
                                            cursor, order, E);
    k_main<<<1024, 128, 0, stream>>>(x, edge_src, edge_dst, order, Wt, cnt, binStart, out);
  } else {
    k_fallback<<<4096, 256, 0, stream>>>(x, node_type, edge_src, edge_dst, edge_type,
                                         weight, w_comp, out, E);
  }
}